// MultiScaleDeformableAttention_20650202759485
// MI455X (gfx1250) — compile-verified
//
#include <hip/hip_runtime.h>
#include <hip/hip_bf16.h>
#include <math.h>

// ---------------- problem constants (match reference) ----------------
#define D_MODEL   256
#define N_HEADS   8
#define HEAD_DIM  32
#define N_LEVELS  4
#define N_POINTS  4
#define BATCH     2
#define LQ        13294          // sum of H*W over levels
#define NQ        (BATCH * LQ)   // 26588 rows for every GEMM

typedef __attribute__((ext_vector_type(2))) float v2f;
typedef __attribute__((ext_vector_type(8))) float v8f;

// =====================================================================
// fp32 WMMA GEMM:  C[M x NCOLS] = A[M x 256] @ W[256 x NCOLS] + bias
//
// Block = 256 threads (8 waves): one 64-column N-tile, 8 32-row M-tiles.
// Each wave owns a 32x64 output tile: 2 A fragments x 4 B fragments ->
// 8 accumulators, so every B ds_load_b64 feeds TWO WMMAs (halves LDS
// traffic per matrix op vs a 16x64 tile).
// The 256x64 weight panel (64 KB) is staged in LDS, K-pair interleaved
// so a lane's B fragment {W[ka][col], W[ka+1][col]} is one ds_load_b64
// (pairs fuse into ds_load_2addr_b64). Unroll x2 overlaps next-iter
// DS/global loads with current WMMAs (separate DScnt / LOADcnt).
// =====================================================================
template <int NCOLS>
__global__ __launch_bounds__(256)
void msda_wmma_gemm_kernel(const float* __restrict__ A,
                           const float* __restrict__ W,
                           const float* __restrict__ bias,
                           float* __restrict__ C,
                           int Mrows) {
    constexpr int TILES_N = NCOLS / 64;

    // LDS weight panel: [kp 0..127][col 0..63][pair 0..1] = 64 KB
    __shared__ float ldsW[D_MODEL * 64];

    const int tileN   = blockIdx.x % TILES_N;
    const int mBlock  = blockIdx.x / TILES_N;
    const int colBase = tileN << 6;

    // ---- cooperative fill of the weight panel -----------------------
    {
        const float* __restrict__ Wp = W + colBase;
        for (int e = threadIdx.x; e < D_MODEL * 64; e += 256) {
            const int row = e >> 6;           // K index (wave-uniform)
            const int col = e & 63;           // lanes vary col -> coalesced
            ldsW[(((row >> 1) * 64 + col) << 1) | (row & 1)] =
                Wp[row * NCOLS + col];
        }
    }
    __syncthreads();

    const int wid  = threadIdx.x >> 5;
    const int lane = threadIdx.x & 31;
    const int half = lane >> 4;               // 0: lanes 0-15, 1: lanes 16-31
    const int idx  = lane & 15;

    const int Mtiles = (Mrows + 31) >> 5;     // 32-row tiles
    const int tileM  = mBlock * 8 + wid;
    if (tileM >= Mtiles) return;              // uniform per-wave exit

    // Two A-fragment rows per lane (rows tileM*32+idx and +16+idx),
    // clamped for the ragged tail (clamped rows are never stored).
    const int rowA0  = tileM * 32 + idx;
    const int rowA1  = rowA0 + 16;
    const int rowA0c = (rowA0 < Mrows) ? rowA0 : (Mrows - 1);
    const int rowA1c = (rowA1 < Mrows) ? rowA1 : (Mrows - 1);
    const float* __restrict__ Arow0 = A + (size_t)rowA0c * D_MODEL + 2 * half;
    const float* __restrict__ Arow1 = A + (size_t)rowA1c * D_MODEL + 2 * half;

    // 8 accumulators init with bias (lane column = colBase + j*16 + idx)
    v8f acc0[4], acc1[4];
#pragma unroll
    for (int j = 0; j < 4; ++j) {
        float bvv = bias[colBase + j * 16 + idx];
        v8f t = bvv;                          // splat
        acc0[j] = t;
        acc1[j] = t;
    }

    const v2f* __restrict__ bpanel = reinterpret_cast<const v2f*>(ldsW);

    // ---- K loop: K = 256, 4 per WMMA, 8 WMMAs per k-step ------------
#pragma unroll 2
    for (int k = 0; k < D_MODEL; k += 4) {
        const int kp = (k >> 1) + half;       // K-pair row in LDS panel

        v2f a0, a1;
        a0.x = Arow0[k];
        a0.y = Arow0[k + 1];
        a1.x = Arow1[k];
        a1.y = Arow1[k + 1];

        v2f b0 = bpanel[kp * 64 +      idx];
        v2f b1 = bpanel[kp * 64 + 16 + idx];
        v2f b2 = bpanel[kp * 64 + 32 + idx];
        v2f b3 = bpanel[kp * 64 + 48 + idx];

        acc0[0] = __builtin_amdgcn_wmma_f32_16x16x4_f32(false, a0, false, b0,
                                                        (short)0, acc0[0], false, false);
        acc1[0] = __builtin_amdgcn_wmma_f32_16x16x4_f32(false, a1, false, b0,
                                                        (short)0, acc1[0], false, false);
        acc0[1] = __builtin_amdgcn_wmma_f32_16x16x4_f32(false, a0, false, b1,
                                                        (short)0, acc0[1], false, false);
        acc1[1] = __builtin_amdgcn_wmma_f32_16x16x4_f32(false, a1, false, b1,
                                                        (short)0, acc1[1], false, false);
        acc0[2] = __builtin_amdgcn_wmma_f32_16x16x4_f32(false, a0, false, b2,
                                                        (short)0, acc0[2], false, false);
        acc1[2] = __builtin_amdgcn_wmma_f32_16x16x4_f32(false, a1, false, b2,
                                                        (short)0, acc1[2], false, false);
        acc0[3] = __builtin_amdgcn_wmma_f32_16x16x4_f32(false, a0, false, b3,
                                                        (short)0, acc0[3], false, false);
        acc1[3] = __builtin_amdgcn_wmma_f32_16x16x4_f32(false, a1, false, b3,
                                                        (short)0, acc1[3], false, false);
    }

    // ---- store epilogue --------------------------------------------
    // acc{h}[j] VGPR r -> row (tileM*32 + h*16 + r + 8*half),
    //                     col (colBase + j*16 + idx).
    const int rowBase0 = tileM * 32 + 8 * half;
    float* __restrict__ Crow0 = C + (size_t)rowBase0 * NCOLS + colBase + idx;
    float* __restrict__ Crow1 = Crow0 + (size_t)16 * NCOLS;

    if ((tileM + 1) * 32 <= Mrows) {
        // fast path: no guards, affine immediate-offset coalesced stores
#pragma unroll
        for (int r = 0; r < 8; ++r) {
#pragma unroll
            for (int j = 0; j < 4; ++j) {
                Crow0[(size_t)r * NCOLS + j * 16] = acc0[j][r];
                Crow1[(size_t)r * NCOLS + j * 16] = acc1[j][r];
            }
        }
    } else {
#pragma unroll
        for (int r = 0; r < 8; ++r) {
            if (rowBase0 + r < Mrows) {
#pragma unroll
                for (int j = 0; j < 4; ++j)
                    Crow0[(size_t)r * NCOLS + j * 16] = acc0[j][r];
            }
            if (rowBase0 + 16 + r < Mrows) {
#pragma unroll
                for (int j = 0; j < 4; ++j)
                    Crow1[(size_t)r * NCOLS + j * 16] = acc1[j][r];
            }
        }
    }
}

// =====================================================================
// Softmax over groups of 16 (L*P) logits, in place.
// =====================================================================
__global__ __launch_bounds__(256)
void msda_softmax16_kernel(float* __restrict__ aw, int rows16) {
    const int r = blockIdx.x * blockDim.x + threadIdx.x;
    if (r >= rows16) return;
    float* p = aw + (size_t)r * 16;

    float v[16];
#pragma unroll
    for (int i = 0; i < 4; ++i)
        reinterpret_cast<float4*>(v)[i] = reinterpret_cast<const float4*>(p)[i];

    float mx = v[0];
#pragma unroll
    for (int i = 1; i < 16; ++i) mx = fmaxf(mx, v[i]);
    float s = 0.f;
#pragma unroll
    for (int i = 0; i < 16; ++i) { v[i] = __expf(v[i] - mx); s += v[i]; }
    const float inv = 1.f / s;
#pragma unroll
    for (int i = 0; i < 16; ++i) v[i] *= inv;

#pragma unroll
    for (int i = 0; i < 4; ++i)
        reinterpret_cast<float4*>(p)[i] = reinterpret_cast<const float4*>(v)[i];
}

// =====================================================================
// Deformable sampling core. One wave32 per (n,lq,head); lane = channel d.
// value gathers are 128B-coalesced ([n,s,m,32] rows); offsets/weights are
// wave-uniform -> uniform branches (grid_sample zeros-padding semantics).
// =====================================================================
__global__ __launch_bounds__(256)
void msda_core_kernel(const float* __restrict__ value,   // [N, S, M, 32]
                      const float* __restrict__ refp,    // [N, Lq, L, 2]
                      const float* __restrict__ off,     // [N, Lq, M, L, P, 2]
                      const float* __restrict__ aw,      // [N, Lq, M, L*P]
                      float* __restrict__ outc,          // [N, Lq, M*32]
                      int nQ) {
    const int wid        = threadIdx.x >> 5;
    const int waveGlobal = blockIdx.x * (blockDim.x >> 5) + wid;
    const int lane       = threadIdx.x & 31;

    const int q = waveGlobal >> 3;          // n*Lq + lq
    const int m = waveGlobal & 7;
    if (q >= nQ) return;
    const int n = q / LQ;

    const int Hs[N_LEVELS] = {100, 50, 25, 13};
    const int Ws[N_LEVELS] = {100, 50, 25, 13};
    const int St[N_LEVELS] = {0, 10000, 12500, 13125};

    const size_t vbase = ((size_t)n * LQ) * D_MODEL + m * HEAD_DIM + lane;

    float acc = 0.f;
#pragma unroll
    for (int l = 0; l < N_LEVELS; ++l) {
        const int   Wl = Ws[l], Hl = Hs[l];
        const float fW = (float)Wl, fH = (float)Hl;
        const float rx = refp[((size_t)q * N_LEVELS + l) * 2 + 0];
        const float ry = refp[((size_t)q * N_LEVELS + l) * 2 + 1];
#pragma unroll
        for (int p = 0; p < N_POINTS; ++p) {
            const size_t ob = ((size_t)q * N_HEADS + m) * 16 + l * N_POINTS + p;
            const float ox = off[ob * 2 + 0];
            const float oy = off[ob * 2 + 1];
            const float a  = aw[ob];

            const float x = rx * fW + ox - 0.5f;   // align_corners=False
            const float y = ry * fH + oy - 0.5f;
            const float x0f = floorf(x), y0f = floorf(y);
            const float wx = x - x0f, wy = y - y0f;
            const int x0 = (int)x0f, y0 = (int)y0f;
            const int x1 = x0 + 1, y1 = y0 + 1;

            const float w00 = (1.f - wx) * (1.f - wy) * a;
            const float w01 = wx * (1.f - wy) * a;
            const float w10 = (1.f - wx) * wy * a;
            const float w11 = wx * wy * a;

            const bool xin0 = (x0 >= 0) & (x0 < Wl);
            const bool xin1 = (x1 >= 0) & (x1 < Wl);

            if ((y0 >= 0) & (y0 < Hl)) {
                const size_t rowb = vbase + (size_t)(St[l] + y0 * Wl) * D_MODEL;
                if (xin0) acc += w00 * value[rowb + (size_t)x0 * D_MODEL];
                if (xin1) acc += w01 * value[rowb + (size_t)x1 * D_MODEL];
            }
            if ((y1 >= 0) & (y1 < Hl)) {
                const size_t rowb = vbase + (size_t)(St[l] + y1 * Wl) * D_MODEL;
                if (xin0) acc += w10 * value[rowb + (size_t)x0 * D_MODEL];
                if (xin1) acc += w11 * value[rowb + (size_t)x1 * D_MODEL];
            }
        }
    }
    outc[(size_t)q * D_MODEL + m * HEAD_DIM + lane] = acc;
}

// =====================================================================
// Launch
// =====================================================================
extern "C" void kernel_launch(void* const* d_in, const int* in_sizes, int n_in,
                              void* d_out, int out_size, void* d_ws, size_t ws_size,
                              hipStream_t stream) {
    const float* query = (const float*)d_in[0];
    const float* refp  = (const float*)d_in[1];
    const float* xin   = (const float*)d_in[2];
    // d_in[3], d_in[4]: spatial shapes / level starts (static, hardcoded)
    const float* Wv    = (const float*)d_in[5];
    const float* bv    = (const float*)d_in[6];
    const float* Woff  = (const float*)d_in[7];
    const float* boff  = (const float*)d_in[8];
    const float* Wattn = (const float*)d_in[9];
    const float* battn = (const float*)d_in[10];
    const float* Wout  = (const float*)d_in[11];
    const float* bout  = (const float*)d_in[12];
    float* out = (float*)d_out;

    // workspace layout (floats)
    float* ws    = (float*)d_ws;
    float* value = ws;                              // NQ * 256
    float* off   = value + (size_t)NQ * 256;        // NQ * 256
    float* aw    = off   + (size_t)NQ * 256;        // NQ * 128
    float* core  = aw    + (size_t)NQ * 128;        // NQ * 256

    const int Mrows   = NQ;                         // 26588
    const int Mtiles  = (Mrows + 31) / 32;          // 831 (32-row tiles)
    const int mBlocks = (Mtiles + 7) / 8;           // 104

    // 1) value = input_flatten @ Wv + bv   (N=256, 4 N-tiles)
    // 2) off   = query @ Woff + boff       (N=256)
    {
        const int blocks = mBlocks * 4;
        msda_wmma_gemm_kernel<256><<<blocks, 256, 0, stream>>>(xin,   Wv,   bv,   value, Mrows);
        msda_wmma_gemm_kernel<256><<<blocks, 256, 0, stream>>>(query, Woff, boff, off,   Mrows);
    }
    // 3) attn logits = query @ Wattn + battn  (N=128, 2 N-tiles)
    {
        const int blocks = mBlocks * 2;
        msda_wmma_gemm_kernel<128><<<blocks, 256, 0, stream>>>(query, Wattn, battn, aw, Mrows);
    }
    // 4) softmax over L*P=16
    {
        const int rows16 = NQ * N_HEADS;            // 212704
        msda_softmax16_kernel<<<(rows16 + 255) / 256, 256, 0, stream>>>(aw, rows16);
    }
    // 5) deformable bilinear sampling core (one wave per (q, head))
    {
        const int waves  = NQ * N_HEADS;
        const int blocks = (waves + 7) / 8;
        msda_core_kernel<<<blocks, 256, 0, stream>>>(value, refp, off, aw, core, NQ);
    }
    // 6) out = core @ Wout + bout  (N=256)
    {
        const int blocks = mBlocks * 4;
        msda_wmma_gemm_kernel<256><<<blocks, 256, 0, stream>>>(core, Wout, bout, out, Mrows);
    }
}